// BatchSparseDenseMatmul_45758581572284
// MI455X (gfx1250) — compile-verified
//
#include <hip/hip_runtime.h>
#include <stdint.h>

// Batched COO SpMM for MI455X (gfx1250, wave32).
// y[b,r] = sum_k vals[k] * x[b, cols[k]] where rows[k]==r.
// Memory/atomic-bound: whole working set (~18 MB) is L2-resident (192 MB L2).
// CDNA5 paths: global_load_async_to_lds_b32 (ASYNCcnt) bulk COO staging,
// clause of hardware global_atomic_add_f32 (L2-side f32 add), wave32 layout
// (32 lanes x float4 == all 128 batch elements per nonzero).

namespace {
constexpr int kB = 128;      // batch
constexpr int kR = 8192;     // rows
constexpr int kC = 16384;    // cols
constexpr int kTile = 32;    // transpose tile
constexpr int kChunk = 512;  // nonzeros staged per block in the scatter kernel
}

// ---------- CDNA5-specific helpers (inline asm per CDNA5 ISA docs) ----------

// Async bulk copy global -> LDS, tracked by ASYNCcnt (CDNA5 ISA, opcode 96).
__device__ __forceinline__ void async_g2l_b32(void* lds_dst, const void* gsrc) {
  asm volatile("global_load_async_to_lds_b32 %0, %1, off"
               :: "v"((unsigned)(uintptr_t)lds_dst),
                  "v"((unsigned long long)(uintptr_t)gsrc)
               : "memory");
}

__device__ __forceinline__ void wait_asynccnt0() {
  asm volatile("s_wait_asynccnt 0" ::: "memory");
}

// Clause of four no-return f32 atomic adds to one 16B-contiguous destination.
// One shared 64-bit address + imm offsets (VFLAT IOFFSET); s_clause 0x3 keeps
// the VMEM pipe on this wave for all four (legal non-flat store clause).
__device__ __forceinline__ void gatomic_add_f32x4(float* p, float a, float b,
                                                  float c, float d) {
  asm volatile("s_clause 0x3\n\t"
               "global_atomic_add_f32 %0, %1, off\n\t"
               "global_atomic_add_f32 %0, %2, off offset:4\n\t"
               "global_atomic_add_f32 %0, %3, off offset:8\n\t"
               "global_atomic_add_f32 %0, %4, off offset:12"
               :: "v"((unsigned long long)(uintptr_t)p),
                  "v"(a), "v"(b), "v"(c), "v"(d));
}

// ---------------------------- kernels ----------------------------

// x[128, 16384] -> xT[16384, 128] so x[:, col] becomes one contiguous 512B row.
__global__ __launch_bounds__(256) void transpose_x_kernel(const float* __restrict__ x,
                                                          float* __restrict__ xT) {
  __shared__ float tile[kTile][kTile + 1];
  const int tx = threadIdx.x, ty = threadIdx.y;
  const int c0 = blockIdx.x * kTile, b0 = blockIdx.y * kTile;
#pragma unroll
  for (int k = 0; k < 4; ++k)
    tile[ty + k * 8][tx] = x[(size_t)(b0 + ty + k * 8) * kC + (c0 + tx)];
  __syncthreads();
#pragma unroll
  for (int k = 0; k < 4; ++k)
    xT[(size_t)(c0 + ty + k * 8) * kB + (b0 + tx)] = tile[tx][ty + k * 8];
}

__global__ __launch_bounds__(256) void zero_kernel(float4* __restrict__ p, int n4) {
  const int i = blockIdx.x * blockDim.x + threadIdx.x;
  if (i < n4) p[i] = make_float4(0.f, 0.f, 0.f, 0.f);
}

// Each block: async-stage kChunk COO triples into LDS, then each wave owns 64
// nonzeros. Per nonzero: broadcast (val,row,col) via readlane, gather one
// float4/lane from xT (coalesced 512B), scatter a 4-atomic clause/lane into yT.
// Tail blocks: out-of-range slots are staged as (val=0,row=0,col=0), which
// contributes atomicAdd(yT[0..127], 0.0f) — a numeric no-op — so the inner
// loop needs NO bounds check (straight-line, clause-friendly body).
__global__ __launch_bounds__(256) void coo_scatter_kernel(
    const float* __restrict__ vals, const int* __restrict__ rows,
    const int* __restrict__ cols, const float* __restrict__ xT,
    float* __restrict__ yT, int nnz) {
  __shared__ unsigned sh[3 * kChunk];  // [vals | rows | cols]
  const int t = threadIdx.x;
  const int base = blockIdx.x * kChunk;

#pragma unroll
  for (int g = 0; g < kChunk / 256; ++g) {
    const int i = g * 256 + t;
    const int gi = base + i;
    if (gi < nnz) {
      async_g2l_b32(&sh[i],              vals + gi);
      async_g2l_b32(&sh[kChunk + i],     rows + gi);
      async_g2l_b32(&sh[2 * kChunk + i], cols + gi);
    } else {
      sh[i] = 0u; sh[kChunk + i] = 0u; sh[2 * kChunk + i] = 0u;
    }
  }
  wait_asynccnt0();   // my wave's async copies landed in LDS
  __syncthreads();    // everyone's copies visible block-wide

  const int wave = t >> 5;
  const int lane = t & 31;
  const float4* xt4 = (const float4*)xT;
  float* const ybase = yT + (lane << 2);   // lane-dependent part, hoisted

#pragma unroll
  for (int g = 0; g < 2; ++g) {
    const int slot = wave * 64 + g * 32 + lane;
    const float mv = __uint_as_float(sh[slot]);
    const int   mr = (int)sh[kChunk + slot];
    const int   mc = (int)sh[2 * kChunk + slot];
#pragma unroll
    for (int j = 0; j < 32; ++j) {
      const float v = __shfl(mv, j, 32);  // literal lane -> v_readlane broadcast
      const int   r = __shfl(mr, j, 32);
      const int   c = __shfl(mc, j, 32);
      const float4 xv = xt4[(size_t)c * (kB / 4) + lane];  // 512B/wave, L2 hit
      gatomic_add_f32x4(ybase + ((size_t)r << 7),          // 512B/wave of atomics
                        v * xv.x, v * xv.y, v * xv.z, v * xv.w);
    }
  }
}

// yT[8192, 128] -> y[128, 8192]
__global__ __launch_bounds__(256) void transpose_y_kernel(const float* __restrict__ yT,
                                                          float* __restrict__ y) {
  __shared__ float tile[kTile][kTile + 1];
  const int tx = threadIdx.x, ty = threadIdx.y;
  const int r0 = blockIdx.x * kTile, b0 = blockIdx.y * kTile;
#pragma unroll
  for (int k = 0; k < 4; ++k)
    tile[ty + k * 8][tx] = yT[(size_t)(r0 + ty + k * 8) * kB + (b0 + tx)];
  __syncthreads();
#pragma unroll
  for (int k = 0; k < 4; ++k)
    y[(size_t)(b0 + ty + k * 8) * kR + (r0 + tx)] = tile[tx][ty + k * 8];
}

// ---------------------------- launcher ----------------------------

extern "C" void kernel_launch(void* const* d_in, const int* in_sizes, int n_in,
                              void* d_out, int out_size, void* d_ws, size_t ws_size,
                              hipStream_t stream) {
  const float* x    = (const float*)d_in[0];  // [B, C]
  const float* vals = (const float*)d_in[1];  // [NNZ]
  const int*   rows = (const int*)d_in[2];    // [NNZ]
  const int*   cols = (const int*)d_in[3];    // [NNZ]
  float* y = (float*)d_out;                   // [B, R]
  const int nnz = in_sizes[1];

  // Workspace: xT (8 MB) + yT (4 MB) = 12 MB, both L2-resident.
  float* xT = (float*)d_ws;
  float* yT = xT + (size_t)kC * kB;

  transpose_x_kernel<<<dim3(kC / kTile, kB / kTile), dim3(32, 8), 0, stream>>>(x, xT);

  const int n4 = kR * kB / 4;
  zero_kernel<<<(n4 + 255) / 256, 256, 0, stream>>>((float4*)yT, n4);

  const int nblk = (nnz + kChunk - 1) / kChunk;  // 1024 blocks of 8 waves
  coo_scatter_kernel<<<nblk, 256, 0, stream>>>(vals, rows, cols, xT, yT, nnz);

  transpose_y_kernel<<<dim3(kR / kTile, kB / kTile), dim3(32, 8), 0, stream>>>(yT, y);
}